// UnifiedGAT_Brain_18081812316372
// MI455X (gfx1250) — compile-verified
//
#include <hip/hip_runtime.h>
#include <math.h>

// ---------------- problem constants ----------------
constexpr int N_   = 25600;
constexpr int E_   = 409600;
constexpr int G_   = 128;
constexpr int IN_  = 200;
constexpr int IN2_ = 400;
constexpr int H_   = 4;
constexpr int C_   = 64;
constexpr int HC_  = 256;

typedef __attribute__((ext_vector_type(2))) float v2f;
typedef __attribute__((ext_vector_type(8))) float v8f;

// ---------------- helpers ----------------
__device__ __forceinline__ float nan_clean(float v) {
  if (isnan(v)) return 0.f;
  if (isinf(v)) return v > 0.f ? 3.4028234663852886e38f : -3.4028234663852886e38f;
  return v;
}

__global__ void fill_f32(float* p, float v, int n) {
  int i = blockIdx.x * blockDim.x + threadIdx.x;
  if (i < n) p[i] = v;
}

// ---------------- preprocessing ----------------
// x2[:, :IN] = nan_to_num(x); x2[:, IN:2IN] = isnan(x)
__global__ void prep_x(const float* __restrict__ x, float* __restrict__ x2) {
  int idx = blockIdx.x * blockDim.x + threadIdx.x;
  if (idx >= N_ * IN_) return;
  int n = idx / IN_, i = idx - n * IN_;
  float v = x[idx];
  int isn = isnan(v);
  x2[(size_t)n * IN2_ + i]       = isn ? 0.f : nan_clean(v);
  x2[(size_t)n * IN2_ + IN_ + i] = isn ? 1.f : 0.f;
}

// clean edge_attr + per-destination degree count & attr sum (self-loop 'mean' fill)
__global__ void prep_ea(const float* __restrict__ ea, const int* __restrict__ ei,
                        float* __restrict__ eac, int* __restrict__ cnt,
                        float* __restrict__ esum) {
  int e = blockIdx.x * blockDim.x + threadIdx.x;
  if (e >= E_) return;
  float v = nan_clean(ea[e]);
  eac[e] = v;
  int d = ei[E_ + e];
  atomicAdd(&cnt[d], 1);
  atomicAdd(&esum[d], v);
}

__global__ void loop_ea_k(const int* __restrict__ cnt, const float* __restrict__ esum,
                          float* __restrict__ lea) {
  int n = blockIdx.x * blockDim.x + threadIdx.x;
  if (n < N_) lea[n] = esum[n] / fmaxf((float)cnt[n], 1.f);
}

// a_e scalar per (layer, head): We is [1, H*C] so a_e[e,h] = ea[e] * dot(We[h,:], ae[h,:])
__global__ void wedot_k(const float* We0, const float* ae0,
                        const float* We1, const float* ae1,
                        const float* We2, const float* ae2, float* wed) {
  int i = threadIdx.x;
  if (i >= 9) return;
  const float* We; const float* ae; int hd;
  if (i < 4)      { We = We0; ae = ae0; hd = i; }
  else if (i < 8) { We = We1; ae = ae1; hd = i - 4; }
  else            { We = We2; ae = ae2; hd = 0; }
  float s = 0.f;
  for (int c = 0; c < C_; ++c) s += We[hd * C_ + c] * ae[hd * C_ + c];
  wed[i] = s;
}

// ---------------- CSR build (edges bucketed by destination) ----------------
// exclusive prefix sum of per-node in-degree; single workgroup, N_/256 = 100 per thread
__global__ void scan_rowptr(const int* __restrict__ cnt, int* __restrict__ rowptr) {
  __shared__ int sums[256];
  const int CH = N_ / 256;
  int t = threadIdx.x;
  int base = t * CH;
  int s = 0;
  for (int i = 0; i < CH; ++i) s += cnt[base + i];
  sums[t] = s;
  __syncthreads();
  for (int off = 1; off < 256; off <<= 1) {
    int v = (t >= off) ? sums[t - off] : 0;
    __syncthreads();
    sums[t] += v;
    __syncthreads();
  }
  int run = (t == 0) ? 0 : sums[t - 1];
  for (int i = 0; i < CH; ++i) { rowptr[base + i] = run; run += cnt[base + i]; }
  if (t == 255) rowptr[N_] = run;
}

__global__ void fill_perm(const int* __restrict__ ei, const int* __restrict__ rowptr,
                          int* __restrict__ fillc, int* __restrict__ perm) {
  int e = blockIdx.x * blockDim.x + threadIdx.x;
  if (e >= E_) return;
  int d = ei[E_ + e];
  int slot = atomicAdd(&fillc[d], 1);
  perm[rowptr[d] + slot] = e;
}

// ---------------- fp32 WMMA GEMM: C[M,Nc] = A[M,K] @ W[K,Nc] ----------------
// W pre-transposed to Wt[Nc,K] so A and B fragments are contiguous b64 loads.
// Wave computes a 16x32 tile (2 accumulators sharing one A fragment); 8 waves/block.
// V_WMMA_F32_16X16X4_F32 layouts: A 16x4: lanes 0-15 M=lane K={0,1}, lanes 16-31 K={2,3};
// B mirrors; C/D: VGPR r -> rows r / r+8 across lane halves.
__global__ __launch_bounds__(256) void gemm_wmma_f32(
    const float* __restrict__ A, const float* __restrict__ Wt, float* __restrict__ Cp,
    int M, int K, int Nc) {
  int wave = threadIdx.x >> 5;
  int lane = threadIdx.x & 31;
  int half = lane >> 4;
  int l16  = lane & 15;
  int row0 = (blockIdx.x * 8 + wave) * 16;
  int col0 = blockIdx.y * 32;

  const float* arow = A  + (size_t)(row0 + l16) * K      + half * 2;
  const float* b0p  = Wt + (size_t)(col0 + l16) * K      + half * 2;
  const float* b1p  = Wt + (size_t)(col0 + 16 + l16) * K + half * 2;
  v8f acc0 = {}, acc1 = {};
  for (int k = 0; k < K; k += 4) {
    v2f a  = *(const v2f*)(arow + k);
    v2f b0 = *(const v2f*)(b0p + k);
    v2f b1 = *(const v2f*)(b1p + k);
    acc0 = __builtin_amdgcn_wmma_f32_16x16x4_f32(false, a, false, b0, (short)0, acc0, false, false);
    acc1 = __builtin_amdgcn_wmma_f32_16x16x4_f32(false, a, false, b1, (short)0, acc1, false, false);
  }
  float* crow = Cp + (size_t)(row0 + half * 8) * Nc + col0 + l16;
#pragma unroll
  for (int r = 0; r < 8; ++r) {
    crow[(size_t)r * Nc]      = acc0[r];
    crow[(size_t)r * Nc + 16] = acc1[r];
  }
}

__global__ void transpose_k(const float* __restrict__ W, float* __restrict__ Wt,
                            int K, int Nc) {
  int idx = blockIdx.x * blockDim.x + threadIdx.x;
  if (idx >= K * Nc) return;
  int k = idx / Nc, n = idx - k * Nc;
  Wt[(size_t)n * K + k] = W[idx];
}

// ---------------- attention ----------------
// per (node, head) source/dest attention dots
__global__ void node_att(const float* __restrict__ hmat, const float* __restrict__ atts,
                         const float* __restrict__ attd, float* __restrict__ aS,
                         float* __restrict__ aD, int Hd, int Cc) {
  int idx = blockIdx.x * blockDim.x + threadIdx.x;
  if (idx >= N_ * Hd) return;
  int n = idx / Hd, hd = idx - n * Hd;
  const float* hp = hmat + (size_t)n * Hd * Cc + hd * Cc;
  float ss = 0.f, dd = 0.f;
  for (int c = 0; c < Cc; ++c) {
    float hv = hp[c];
    ss += hv * atts[hd * Cc + c];
    dd += hv * attd[hd * Cc + c];
  }
  aS[idx] = ss;
  aD[idx] = dd;
}

// Fused per-destination softmax + aggregation: one wave per node, zero atomics.
// out[n] = (sum_e ex_e * h[src_e]) / (sum_e ex_e + 1e-16), ex_e = exp(alpha_e - max_e alpha)
// Each lane owns CPL = HD*CC/32 consecutive channels (head-aligned: CC/CPL divides 32).
template <int HD, int CC>
__global__ __launch_bounds__(256) void gat_node(
    const int* __restrict__ ei, const float* __restrict__ eac,
    const float* __restrict__ lea, const float* __restrict__ aS,
    const float* __restrict__ aD, const float* __restrict__ wd,
    const int* __restrict__ rowptr, const int* __restrict__ perm,
    const float* __restrict__ hmat, float* __restrict__ outp) {
  constexpr int HCL = HD * CC;
  constexpr int CPL = HCL / 32;
  int wid = (blockIdx.x * blockDim.x + threadIdx.x) >> 5;
  if (wid >= N_) return;
  int lane = threadIdx.x & 31;
  int c0 = lane * CPL;
  int hd = c0 / CC;

  int start = rowptr[wid];
  int end   = rowptr[wid + 1];
  float adn = aD[wid * HD + hd];
  float wdh = wd[hd];

  // self-loop alpha (src = dst = wid, edge_attr = mean of incoming)
  float sa = aS[wid * HD + hd] + adn + lea[wid] * wdh;
  sa = sa > 0.f ? sa : 0.2f * sa;

  // pass 1: neighborhood max (order-independent)
  float m = sa;
  for (int p = start; p < end; ++p) {
    int e = perm[p];
    int s = ei[e];
    float a = aS[s * HD + hd] + adn + eac[e] * wdh;
    a = a > 0.f ? a : 0.2f * a;
    m = fmaxf(m, a);
  }

  // pass 2: exp-sum + weighted accumulation (softmax denominator factored out)
  float acc[CPL];
  float ex = expf(sa - m);
  float z = ex;
  const float* hn = hmat + (size_t)wid * HCL + c0;
#pragma unroll
  for (int i = 0; i < CPL; ++i) acc[i] = hn[i] * ex;
  for (int p = start; p < end; ++p) {
    int e = perm[p];
    int s = ei[e];
    float a = aS[s * HD + hd] + adn + eac[e] * wdh;
    a = a > 0.f ? a : 0.2f * a;
    ex = expf(a - m);
    z += ex;
    const float* hs = hmat + (size_t)s * HCL + c0;
#pragma unroll
    for (int i = 0; i < CPL; ++i) acc[i] += hs[i] * ex;
  }
  float inv = 1.f / (z + 1e-16f);
  float* od = outp + (size_t)wid * HCL + c0;
#pragma unroll
  for (int i = 0; i < CPL; ++i) od[i] = acc[i] * inv;
}

// ---------------- post ops ----------------
__global__ void bias_bn_relu(const float* __restrict__ xc, const float* __restrict__ b,
                             const float* __restrict__ g, const float* __restrict__ be,
                             const float* __restrict__ mn, const float* __restrict__ vr,
                             float* __restrict__ xa) {
  int idx = blockIdx.x * blockDim.x + threadIdx.x;
  if (idx >= N_ * HC_) return;
  int j = idx & (HC_ - 1);
  float v = xc[idx] + b[j];
  v = (v - mn[j]) * rsqrtf(vr[j] + 1e-5f) * g[j] + be[j];
  xa[idx] = fmaxf(v, 0.f);
}

__global__ void bias_only(const float* __restrict__ xc, const float* __restrict__ b,
                          float* __restrict__ xa, int cols, int n) {
  int idx = blockIdx.x * blockDim.x + threadIdx.x;
  if (idx < n) xa[idx] = xc[idx] + b[idx % cols];
}

__global__ void pool_acc(const float* __restrict__ xa, const int* __restrict__ batch,
                         float* __restrict__ gsum, float* __restrict__ gcnt) {
  int idx = blockIdx.x * blockDim.x + threadIdx.x;
  if (idx >= N_ * C_) return;
  int n = idx / C_, j = idx - n * C_;
  int gi = batch[n];
  atomicAdd(&gsum[gi * C_ + j], xa[idx]);
  if (j == 0) atomicAdd(&gcnt[gi], 1.f);
}

__global__ void pool_div(const float* __restrict__ gsum, const float* __restrict__ gcnt,
                         float* __restrict__ gp) {
  int idx = blockIdx.x * blockDim.x + threadIdx.x;
  if (idx >= G_ * C_) return;
  gp[idx] = gsum[idx] / fmaxf(gcnt[idx / C_], 1.f);
}

__global__ void mlp_hidden(const float* __restrict__ gp, const float* __restrict__ w1,
                           const float* __restrict__ b1, float* __restrict__ hid) {
  int idx = blockIdx.x * blockDim.x + threadIdx.x;
  if (idx >= G_ * C_) return;
  int gi = idx / C_, j = idx - gi * C_;
  float s = b1[j];
  for (int c = 0; c < C_; ++c) s += gp[gi * C_ + c] * w1[c * C_ + j];
  hid[idx] = 0.5f * s * (1.f + erff(s * 0.70710678118654752f));  // exact gelu
}

__global__ void mlp_out(const float* __restrict__ hid, const float* __restrict__ w2,
                        const float* __restrict__ b2, float* __restrict__ out) {
  int idx = blockIdx.x * blockDim.x + threadIdx.x;
  if (idx >= G_ * 2) return;
  int gi = idx >> 1, o = idx & 1;
  float s = b2[o];
  for (int c = 0; c < C_; ++c) s += hid[gi * C_ + c] * w2[c * 2 + o];
  out[idx] = s;
}

// ---------------- host launch ----------------
extern "C" void kernel_launch(void* const* d_in, const int* in_sizes, int n_in,
                              void* d_out, int out_size, void* d_ws, size_t ws_size,
                              hipStream_t stream) {
  (void)in_sizes; (void)n_in; (void)out_size; (void)ws_size;
  const float* x     = (const float*)d_in[0];
  const int*   ei    = (const int*)d_in[1];
  const float* ea    = (const float*)d_in[2];
  const int*   batch = (const int*)d_in[3];
  const float* Wl[3]  = {(const float*)d_in[4],  (const float*)d_in[10], (const float*)d_in[16]};
  const float* asl[3] = {(const float*)d_in[5],  (const float*)d_in[11], (const float*)d_in[17]};
  const float* adl[3] = {(const float*)d_in[6],  (const float*)d_in[12], (const float*)d_in[18]};
  const float* Wel[3] = {(const float*)d_in[7],  (const float*)d_in[13], (const float*)d_in[19]};
  const float* ael[3] = {(const float*)d_in[8],  (const float*)d_in[14], (const float*)d_in[20]};
  const float* bl[3]  = {(const float*)d_in[9],  (const float*)d_in[15], (const float*)d_in[21]};
  const float* bng = (const float*)d_in[22];
  const float* bnb = (const float*)d_in[23];
  const float* bnm = (const float*)d_in[24];
  const float* bnv = (const float*)d_in[25];
  const float* cw1 = (const float*)d_in[26];
  const float* cb1 = (const float*)d_in[27];
  const float* cw2 = (const float*)d_in[28];
  const float* cb2 = (const float*)d_in[29];
  float* out = (float*)d_out;

  // workspace carve-out (~100 MB)
  float* ws = (float*)d_ws;
  size_t o = 0;
  auto carve = [&](size_t n) { float* p = ws + o; o += n; return p; };
  float* xA     = carve((size_t)N_ * IN2_);   // layer input (400 wide L0, 256 after)
  float* xB     = carve((size_t)N_ * HC_);    // GEMM output h
  float* xC     = carve((size_t)N_ * HC_);    // aggregated output
  float* aS     = carve((size_t)N_ * H_);
  float* aD     = carve((size_t)N_ * H_);
  float* esum   = carve(N_);
  float* lea    = carve(N_);
  float* eac    = carve(E_);
  float* wed    = carve(16);
  float* Wt     = carve((size_t)HC_ * IN2_);  // transposed weights (max 256x400)
  int*   cnti   = (int*)carve(N_);
  int*   rowptr = (int*)carve(N_ + 1);
  int*   fillc  = (int*)carve(N_);
  int*   perm   = (int*)carve(E_);
  float* gsum   = carve((size_t)G_ * C_);
  float* gcnt   = carve(G_);
  float* gp     = carve((size_t)G_ * C_);
  float* hid    = carve((size_t)G_ * C_);

  auto blks = [](long n, int b) { return (unsigned)((n + b - 1) / b); };
  const int T = 256;

  // ---- preprocess + CSR build ----
  fill_f32<<<blks(N_, T), T, 0, stream>>>((float*)cnti, 0.f, N_);
  fill_f32<<<blks(N_, T), T, 0, stream>>>(esum, 0.f, N_);
  fill_f32<<<blks(N_, T), T, 0, stream>>>((float*)fillc, 0.f, N_);
  prep_x<<<blks((long)N_ * IN_, T), T, 0, stream>>>(x, xA);
  prep_ea<<<blks(E_, T), T, 0, stream>>>(ea, ei, eac, cnti, esum);
  loop_ea_k<<<blks(N_, T), T, 0, stream>>>(cnti, esum, lea);
  wedot_k<<<1, 32, 0, stream>>>(Wel[0], ael[0], Wel[1], ael[1], Wel[2], ael[2], wed);
  scan_rowptr<<<1, 256, 0, stream>>>(cnti, rowptr);
  fill_perm<<<blks(E_, T), T, 0, stream>>>(ei, rowptr, fillc, perm);

  // ---- 3 GAT layers ----
  for (int l = 0; l < 3; ++l) {
    int din = (l == 0) ? IN2_ : HC_;
    int Hd  = (l < 2) ? H_ : 1;
    int HCl = Hd * C_;
    const float* wd = wed + ((l == 0) ? 0 : (l == 1) ? 4 : 8);

    transpose_k<<<blks((long)din * HCl, T), T, 0, stream>>>(Wl[l], Wt, din, HCl);
    dim3 gg(N_ / 128, HCl / 32);
    gemm_wmma_f32<<<gg, 256, 0, stream>>>(xA, Wt, xB, N_, din, HCl);

    node_att<<<blks((long)N_ * Hd, T), T, 0, stream>>>(xB, asl[l], adl[l], aS, aD, Hd, C_);
    if (l < 2)
      gat_node<4, 64><<<blks((long)N_ * 32, T), T, 0, stream>>>(ei, eac, lea, aS, aD, wd,
                                                                rowptr, perm, xB, xC);
    else
      gat_node<1, 64><<<blks((long)N_ * 32, T), T, 0, stream>>>(ei, eac, lea, aS, aD, wd,
                                                                rowptr, perm, xB, xC);
    if (l < 2)
      bias_bn_relu<<<blks((long)N_ * HC_, T), T, 0, stream>>>(xC, bl[l], bng, bnb, bnm, bnv, xA);
    else
      bias_only<<<blks((long)N_ * C_, T), T, 0, stream>>>(xC, bl[l], xA, C_, N_ * C_);
  }

  // ---- mean pool + MLP head ----
  fill_f32<<<blks((long)G_ * C_, T), T, 0, stream>>>(gsum, 0.f, G_ * C_);
  fill_f32<<<blks(G_, T), T, 0, stream>>>(gcnt, 0.f, G_);
  pool_acc<<<blks((long)N_ * C_, T), T, 0, stream>>>(xA, batch, gsum, gcnt);
  pool_div<<<blks((long)G_ * C_, T), T, 0, stream>>>(gsum, gcnt, gp);
  mlp_hidden<<<blks((long)G_ * C_, T), T, 0, stream>>>(gp, cw1, cb1, hid);
  mlp_out<<<blks(G_ * 2, T), T, 0, stream>>>(hid, cw2, cb2, out);
}